// MPNN_75058848465162
// MI455X (gfx1250) — compile-verified
//
#include <hip/hip_runtime.h>

#define N_NODES 50000
#define DIM 128
#define MTILES (N_NODES / 16)   // 3125
#define OTILES (DIM / 16)       // 8

typedef __attribute__((ext_vector_type(2))) float v2f;
typedef __attribute__((ext_vector_type(8))) float v8f;

// ---------------------------------------------------------------------------
// Zero the aggregation buffer (float4-vectorized stores).
// ---------------------------------------------------------------------------
__global__ void mpnn_zero_f4(float4* __restrict__ p, int n4) {
  int i = blockIdx.x * blockDim.x + threadIdx.x;
  if (i < n4) p[i] = make_float4(0.f, 0.f, 0.f, 0.f);
}

// ---------------------------------------------------------------------------
// Edge aggregation: one wave32 per edge. Lane l moves floats [4l, 4l+4) of the
// 128-wide feature row: float4 gather from feat[src[e]], f32 atomic scatter to
// agg[dst[e]]. Working set (25.6 MB table + 25.6 MB accumulator) is L2
// resident on MI455X (192 MB L2), so this runs at L2/atomic-unit rate.
// ---------------------------------------------------------------------------
__global__ void mpnn_edge_scatter(const float* __restrict__ feat,
                                  const long long* __restrict__ src,
                                  const long long* __restrict__ dst,
                                  float* __restrict__ agg,
                                  int n_edges) {
  int e = blockIdx.x * (blockDim.x >> 5) + (threadIdx.x >> 5);
  if (e >= n_edges) return;
  int lane = threadIdx.x & 31;
  long long s = src[e];
  long long d = dst[e];
  float4 v = ((const float4*)(feat + s * (long long)DIM))[lane];
  float* o = agg + d * (long long)DIM + lane * 4;
  atomicAdd(o + 0, v.x);
  atomicAdd(o + 1, v.y);
  atomicAdd(o + 2, v.z);
  atomicAdd(o + 3, v.w);
}

// ---------------------------------------------------------------------------
// out[n, o] = act( sum_k A[n,k] * W[o,k] + bias[o] )  via V_WMMA_F32_16X16X4_F32
// One wave per 16x16 output tile, 32 WMMA steps over K=128.
// Fragment layouts per CDNA5 ISA 7.12.2:
//   A 16x4 f32 : lane<16 -> row mBase+lane, K={k0,k0+1}; lane>=16 -> K={k0+2,k0+3}
//   B 4x16 f32 : lane<16 -> col oBase+lane, K={k0,k0+1}; lane>=16 -> K={k0+2,k0+3}
//     (B[k][o] = W[o][k], so each lane reads consecutive elems of its W row)
//   C/D 16x16 f32: VGPR v, lane<16 -> [M=v][N=lane]; lane>=16 -> [M=v+8][N=lane-16]
// ---------------------------------------------------------------------------
__global__ void mpnn_gemm_bias_act(const float* __restrict__ A,   // [N_NODES][DIM]
                                   const float* __restrict__ W,   // [DIM out][DIM in]
                                   const float* __restrict__ bias,// [DIM]
                                   float* __restrict__ out,       // [N_NODES][DIM]
                                   int relu) {
  int waveId = blockIdx.x * (blockDim.x >> 5) + (threadIdx.x >> 5);
  int mTile = waveId / OTILES;
  int oTile = waveId % OTILES;
  if (mTile >= MTILES) return;           // wave-uniform: EXEC stays all-ones

  int lane = threadIdx.x & 31;
  int half = lane >> 4;                  // 0: K pair {0,1}; 1: K pair {2,3}
  int l16  = lane & 15;
  int mBase = mTile * 16;
  int oBase = oTile * 16;

  const float* Arow = A + (long long)(mBase + l16) * DIM;  // this lane's A row
  const float* Wrow = W + (long long)(oBase + l16) * DIM;  // this lane's B column (= W row)

  v8f c = {0.f, 0.f, 0.f, 0.f, 0.f, 0.f, 0.f, 0.f};
#pragma unroll
  for (int k0 = 0; k0 < DIM; k0 += 4) {
    int ka = k0 + half * 2;
    v2f a, b;
    a.x = Arow[ka];
    a.y = Arow[ka + 1];
    b.x = Wrow[ka];
    b.y = Wrow[ka + 1];
    // 8 args: (neg_a, A, neg_b, B, c_mod, C, reuse_a, reuse_b)
    c = __builtin_amdgcn_wmma_f32_16x16x4_f32(false, a, false, b,
                                              (short)0, c, false, false);
  }

  float bv = bias[oBase + l16];
#pragma unroll
  for (int v = 0; v < 8; ++v) {
    int m = mBase + v + half * 8;
    float val = c[v] + bv;
    if (relu) val = fmaxf(val, 0.f);
    out[(long long)m * DIM + oBase + l16] = val;
  }
}

// ---------------------------------------------------------------------------
// Launch: zero agg -> scatter(features) -> GEMM1(+ReLU) into d_out (as h) ->
//         zero agg -> scatter(h=d_out)  -> GEMM2 into d_out.
// Workspace use: single N_NODES*DIM float buffer (25.6 MB).
// ---------------------------------------------------------------------------
extern "C" void kernel_launch(void* const* d_in, const int* in_sizes, int n_in,
                              void* d_out, int out_size, void* d_ws, size_t ws_size,
                              hipStream_t stream) {
  const float*     feat = (const float*)d_in[0];
  const long long* src  = (const long long*)d_in[1];
  const long long* dst  = (const long long*)d_in[2];
  const float*     W1   = (const float*)d_in[3];
  const float*     b1   = (const float*)d_in[4];
  const float*     W2   = (const float*)d_in[5];
  const float*     b2   = (const float*)d_in[6];
  float* out = (float*)d_out;
  int n_edges = in_sizes[1];

  float* agg = (float*)d_ws;             // [N_NODES][DIM] accumulator

  const int n4        = N_NODES * DIM / 4;
  const int zeroGrid  = (n4 + 255) / 256;
  const int edgeGrid  = (n_edges + 7) / 8;         // 8 waves (edges) per block
  const int tiles     = MTILES * OTILES;           // 25000 wave-tiles
  const int gemmGrid  = (tiles + 7) / 8;           // 8 waves per block

  // Layer 1: h = relu(segsum(feat) @ W1^T + b1), h stored in d_out
  mpnn_zero_f4<<<zeroGrid, 256, 0, stream>>>((float4*)agg, n4);
  mpnn_edge_scatter<<<edgeGrid, 256, 0, stream>>>(feat, src, dst, agg, n_edges);
  mpnn_gemm_bias_act<<<gemmGrid, 256, 0, stream>>>(agg, W1, b1, out, 1);

  // Layer 2: out = segsum(h) @ W2^T + b2
  mpnn_zero_f4<<<zeroGrid, 256, 0, stream>>>((float4*)agg, n4);
  mpnn_edge_scatter<<<edgeGrid, 256, 0, stream>>>(out, src, dst, agg, n_edges);
  mpnn_gemm_bias_act<<<gemmGrid, 256, 0, stream>>>(agg, W2, b2, out, 0);
}